// MaskRCNN_17609365914120
// MI455X (gfx1250) — compile-verified
//
#include <hip/hip_runtime.h>
#include <stdint.h>

#define B_      2
#define DD      64
#define HH      96
#define WW      96
#define NPB     (DD*HH*WW)        // 589824 boxes per batch
#define TOTAL   (B_*NPB)          // 1179648
#define K_      2048
#define CANDMAX 4096
#define ANCHOR_ 12.0f
#define THR_    0.25f

typedef __attribute__((ext_vector_type(2))) float v2f;
typedef __attribute__((ext_vector_type(8))) float v8f;

// ---- workspace layout (bytes) ----
static constexpr size_t OFF_HIST1  = 0;                    // 2*65536*4 = 524288
static constexpr size_t OFF_HIST2  = 524288;               // 524288
static constexpr size_t OFF_CNT    = 1048576;              // 2*4 (padded)
static constexpr size_t OFF_SEL1   = 1048832;              // 2*2*4 (padded)
static constexpr size_t OFF_SELK   = 1049088;              // 2*4 (padded)
static constexpr size_t OFF_CAND   = 1049600;              // 2*4096*8 = 65536
static constexpr size_t OFF_ZEND   = OFF_CAND + 65536;     // 1115136 (zeroed up to here)
static constexpr size_t OFF_SCORES = 1115136;              // 2*2048*4
static constexpr size_t OFF_BOXES  = 1131520;              // 2*2048*6*4
static constexpr size_t OFF_VOLS   = 1229824;              // 2*2048*4
static constexpr size_t OFF_MASK   = 1246208;              // 2*2048*64*4 = 1048576

__device__ __forceinline__ unsigned orderKey(float f) {
  unsigned u = __float_as_uint(f);
  return (u & 0x80000000u) ? ~u : (u | 0x80000000u);   // monotonic total order
}
__device__ __forceinline__ float unorderKey(unsigned k) {
  unsigned u = (k & 0x80000000u) ? (k ^ 0x80000000u) : ~k;
  return __uint_as_float(u);
}

__global__ void k_zero(unsigned* p, int n) {
  int i = blockIdx.x * 256 + threadIdx.x;
  if (i < n) p[i] = 0u;
}

__global__ void k_hist_hi(const float* __restrict__ scores, unsigned* __restrict__ hist) {
  int g = blockIdx.x * 256 + threadIdx.x;
  if (g >= TOTAL) return;
  int b = (g >= NPB) ? 1 : 0;
  unsigned key = orderKey(scores[g]);
  atomicAdd(&hist[b * 65536 + (key >> 16)], 1u);
}

__global__ void __launch_bounds__(1024) k_sel1(const unsigned* __restrict__ hist,
                                               unsigned* __restrict__ sel1) {
  int b = blockIdx.x, t = threadIdx.x;
  __shared__ unsigned part[1024];
  const unsigned* h = hist + b * 65536;
  unsigned s = 0;
  int base = t * 64;
  for (int j = 0; j < 64; ++j) s += h[base + j];
  part[t] = s;
  __syncthreads();
  if (t == 0) {
    unsigned cum = 0; int g = 1023;
    for (; g > 0; --g) { if (cum + part[g] >= (unsigned)K_) break; cum += part[g]; }
    unsigned T = (unsigned)(g * 64), C = cum;
    for (int d = g * 64 + 63; d >= g * 64; --d) {
      unsigned hv = h[d];
      if (C + hv >= (unsigned)K_) { T = (unsigned)d; break; }
      C += hv;
    }
    sel1[b * 2] = T;       // top-16-bit threshold digit
    sel1[b * 2 + 1] = C;   // count strictly greater (by top digit)
  }
}

__global__ void k_hist_lo(const float* __restrict__ scores, const unsigned* __restrict__ sel1,
                          unsigned* __restrict__ hist2) {
  int g = blockIdx.x * 256 + threadIdx.x;
  if (g >= TOTAL) return;
  int b = (g >= NPB) ? 1 : 0;
  unsigned key = orderKey(scores[g]);
  if ((key >> 16) == sel1[b * 2])
    atomicAdd(&hist2[b * 65536 + (key & 0xFFFFu)], 1u);
}

__global__ void __launch_bounds__(1024) k_sel2(const unsigned* __restrict__ hist2,
                                               const unsigned* __restrict__ sel1,
                                               unsigned* __restrict__ selK) {
  int b = blockIdx.x, t = threadIdx.x;
  __shared__ unsigned part[1024];
  const unsigned* h = hist2 + b * 65536;
  unsigned Ktgt = (unsigned)K_ - sel1[b * 2 + 1];
  unsigned s = 0;
  int base = t * 64;
  for (int j = 0; j < 64; ++j) s += h[base + j];
  part[t] = s;
  __syncthreads();
  if (t == 0) {
    unsigned cum = 0; int g = 1023;
    for (; g > 0; --g) { if (cum + part[g] >= Ktgt) break; cum += part[g]; }
    unsigned T = (unsigned)(g * 64), C = cum;
    for (int d = g * 64 + 63; d >= g * 64; --d) {
      unsigned hv = h[d];
      if (C + hv >= Ktgt) { T = (unsigned)d; break; }
      C += hv;
    }
    selK[b] = (sel1[b * 2] << 16) | T;   // exact 32-bit threshold key
  }
}

__global__ void k_gather(const float* __restrict__ scores, const unsigned* __restrict__ selK,
                         unsigned* __restrict__ cnt, unsigned long long* __restrict__ cand) {
  int g = blockIdx.x * 256 + threadIdx.x;
  if (g >= TOTAL) return;
  int b = (g >= NPB) ? 1 : 0;
  unsigned key = orderKey(scores[g]);
  if (key >= selK[b]) {
    unsigned pos = atomicAdd(&cnt[b], 1u);
    if (pos < CANDMAX) {
      unsigned i = (unsigned)(g - b * NPB);
      // sort descending => (key desc, index asc) via ~idx in low word
      cand[b * CANDMAX + pos] =
          ((unsigned long long)key << 32) | (unsigned long long)(0xFFFFFFFFu - i);
    }
  }
}

// bitonic sort of 4096 candidates in LDS, then deparametrize the 2048 winners
__global__ void __launch_bounds__(1024) k_sortprep(const unsigned long long* __restrict__ cand,
                                                   const float* __restrict__ bboxes,
                                                   float* __restrict__ scoresTop,
                                                   float* __restrict__ boxesTop,
                                                   float* __restrict__ vols) {
  int b = blockIdx.x, t = threadIdx.x;
  __shared__ unsigned long long s[CANDMAX];
  for (int i = t; i < CANDMAX; i += 1024) s[i] = cand[b * CANDMAX + i];
  __syncthreads();
  for (int k = 2; k <= CANDMAX; k <<= 1) {
    for (int j = k >> 1; j > 0; j >>= 1) {
      for (int t2 = t; t2 < CANDMAX / 2; t2 += 1024) {
        int i = ((t2 & ~(j - 1)) << 1) | (t2 & (j - 1));
        int p = i + j;
        unsigned long long a = s[i], c = s[p];
        bool desc = ((i & k) == 0);
        bool sw = desc ? (a < c) : (a > c);
        if (sw) { s[i] = c; s[p] = a; }
      }
      __syncthreads();
    }
  }
  for (int q = t; q < K_; q += 1024) {
    unsigned long long comp = s[q];
    float score = 0.f, bz = 0.f, by = 0.f, bx = 0.f, bd = 0.f, bh = 0.f, bw = 0.f, vol = 1.0f;
    if (comp != 0ull) {
      unsigned key = (unsigned)(comp >> 32);
      unsigned idx = 0xFFFFFFFFu - (unsigned)(comp & 0xFFFFFFFFu);
      score = unorderKey(key);
      int z = (int)(idx / (HH * WW));
      int r = (int)idx - z * (HH * WW);
      int y = r / WW;
      int x = r - y * WW;
      const float* p0 = bboxes + (size_t)b * 6 * NPB + idx;
      float pz = p0[0], py = p0[(size_t)NPB], px = p0[(size_t)2 * NPB];
      float pd = p0[(size_t)3 * NPB], ph = p0[(size_t)4 * NPB], pw = p0[(size_t)5 * NPB];
      bz = pz * ANCHOR_ + (float)z + 0.5f;
      by = py * ANCHOR_ + (float)y + 0.5f;
      bx = px * ANCHOR_ + (float)x + 0.5f;
      bd = expf(pd) * ANCHOR_;
      bh = expf(ph) * ANCHOR_;
      bw = expf(pw) * ANCHOR_;
      vol = bd * bh * bw;
    }
    scoresTop[b * K_ + q] = score;
    float* bt = boxesTop + (size_t)(b * K_ + q) * 6;
    bt[0] = bz; bt[1] = by; bt[2] = bx; bt[3] = bd; bt[4] = bh; bt[5] = bw;
    vols[b * K_ + q] = vol;
  }
}

__device__ __forceinline__ float interVol(float rc0, float rc1, float rc2,
                                          float rs0, float rs1, float rs2,
                                          float cc0, float cc1, float cc2,
                                          float cs0, float cs1, float cs2) {
  float lo0 = fmaxf(rc0 - rs0 * 0.5f, cc0 - cs0 * 0.5f);
  float hi0 = fminf(rc0 + rs0 * 0.5f, cc0 + cs0 * 0.5f);
  float lo1 = fmaxf(rc1 - rs1 * 0.5f, cc1 - cs1 * 0.5f);
  float hi1 = fminf(rc1 + rs1 * 0.5f, cc1 + cs1 * 0.5f);
  float lo2 = fmaxf(rc2 - rs2 * 0.5f, cc2 - cs2 * 0.5f);
  float hi2 = fminf(rc2 + rs2 * 0.5f, cc2 + cs2 * 0.5f);
  float e0 = fmaxf(hi0 - lo0, 0.f);
  float e1 = fmaxf(hi1 - lo1, 0.f);
  float e2 = fmaxf(hi2 - lo2, 0.f);
  return e0 * e1 * e2;
}

// One wave32 per 16-row x 32-col tile. union base vol_i + vol_j via f32 WMMA.
__global__ void __launch_bounds__(32) k_iou(const float* __restrict__ boxesTop,
                                            const float* __restrict__ vols,
                                            unsigned* __restrict__ mask) {
  int w = blockIdx.x;      // 32-column word [0,64)
  int rt = blockIdx.y;     // 16-row tile   [0,128)
  int b = blockIdx.z;
  int lane = threadIdx.x;
  __shared__ float rb[16][8];
  if (lane < 16) {
    int r = rt * 16 + lane;
    const float* bp = boxesTop + (size_t)(b * K_ + r) * 6;
#pragma unroll
    for (int c = 0; c < 6; ++c) rb[lane][c] = bp[c];
    rb[lane][6] = vols[b * K_ + r];
  }
  __syncthreads();

  int n = lane & 15;
  int j0 = w * 32 + n, j1 = j0 + 16;
  const float* c0p = boxesTop + (size_t)(b * K_ + j0) * 6;
  const float* c1p = boxesTop + (size_t)(b * K_ + j1) * 6;
  float c00 = c0p[0], c01 = c0p[1], c02 = c0p[2], c03 = c0p[3], c04 = c0p[4], c05 = c0p[5];
  float c10 = c1p[0], c11 = c1p[1], c12 = c1p[2], c13 = c1p[3], c14 = c1p[4], c15 = c1p[5];
  float v0 = vols[b * K_ + j0];
  float v1 = vols[b * K_ + j1];

  bool lo16 = lane < 16;
  // A (16x4, f32): row m = [vol_m, 1, 0, 0]; lanes 0-15 hold K=0,1; lanes 16-31 K=2,3 (=0)
  v2f A;  A.x = lo16 ? rb[lane][6] : 0.f;  A.y = lo16 ? 1.f : 0.f;
  // B (4x16, f32): col n = [1, vol_n, 0, 0]^T
  v2f B0; B0.x = lo16 ? 1.f : 0.f; B0.y = lo16 ? v0 : 0.f;
  v2f B1; B1.x = lo16 ? 1.f : 0.f; B1.y = lo16 ? v1 : 0.f;
  v8f cz = {};
  // D[m][n] = vol_m + vol_n  (exact: two exact products, one RNE add)
  v8f U0 = __builtin_amdgcn_wmma_f32_16x16x4_f32(false, A, false, B0, (short)0, cz, false, false);
  v8f U1 = __builtin_amdgcn_wmma_f32_16x16x4_f32(false, A, false, B1, (short)0, cz, false, false);

#pragma unroll
  for (int v = 0; v < 8; ++v) {
    int M = lo16 ? v : v + 8;            // D layout: VGPR v holds rows v (lanes 0-15) and v+8
    int i = rt * 16 + M;
    float rc0 = rb[M][0], rc1 = rb[M][1], rc2 = rb[M][2];
    float rs0 = rb[M][3], rs1 = rb[M][4], rs2 = rb[M][5];
    float in0 = interVol(rc0, rc1, rc2, rs0, rs1, rs2, c00, c01, c02, c03, c04, c05);
    float in1 = interVol(rc0, rc1, rc2, rs0, rs1, rs2, c10, c11, c12, c13, c14, c15);
    // iou > thr  <=>  inter > thr * union  (union = volsum - inter > 0)
    bool p0 = (in0 > THR_ * (U0[v] - in0)) && (j0 > i);
    bool p1 = (in1 > THR_ * (U1[v] - in1)) && (j1 > i);
    unsigned bl0 = __builtin_amdgcn_ballot_w32(p0);
    unsigned bl1 = __builtin_amdgcn_ballot_w32(p1);
    if (lane == 0) {
      unsigned rowA = (unsigned)(rt * 16 + v), rowB = rowA + 8;
      mask[((size_t)b * K_ + rowA) * 64 + w] = (bl0 & 0xFFFFu) | (bl1 << 16);
      mask[((size_t)b * K_ + rowB) * 64 + w] = (bl0 >> 16) | (bl1 & 0xFFFF0000u);
    }
  }
}

// sequential greedy scan (exact reference semantics) + output write
__global__ void __launch_bounds__(64) k_nmsout(const unsigned* __restrict__ mask,
                                               const float* __restrict__ scoresTop,
                                               const float* __restrict__ boxesTop,
                                               float* __restrict__ out) {
  int b = blockIdx.x, t = threadIdx.x;
  __shared__ unsigned keep[64];
  keep[t] = 0xFFFFFFFFu;
  __syncthreads();
  for (int i = 0; i < K_; ++i) {
    bool on = (keep[i >> 5] >> (i & 31)) & 1u;
    __syncthreads();
    if (on) keep[t] &= ~mask[((size_t)b * K_ + i) * 64 + t];
    __syncthreads();
  }
  for (int k = t; k < K_; k += 64) {
    bool kp = (keep[k >> 5] >> (k & 31)) & 1u;
    size_t base = (size_t)(b * K_ + k) * 7;
    out[base] = kp ? scoresTop[b * K_ + k] : 0.f;
    const float* bt = boxesTop + (size_t)(b * K_ + k) * 6;
#pragma unroll
    for (int c = 0; c < 6; ++c) out[base + 1 + c] = kp ? bt[c] : 0.f;
  }
}

extern "C" void kernel_launch(void* const* d_in, const int* in_sizes, int n_in,
                              void* d_out, int out_size, void* d_ws, size_t ws_size,
                              hipStream_t stream) {
  (void)in_sizes; (void)n_in; (void)out_size; (void)ws_size;
  const float* bboxes = (const float*)d_in[0];   // (2,6,64,96,96) f32
  const float* scores = (const float*)d_in[1];   // (2,64,96,96) f32
  float* out = (float*)d_out;                    // (2,2048,7) f32
  char* ws = (char*)d_ws;

  unsigned* hist1 = (unsigned*)(ws + OFF_HIST1);
  unsigned* hist2 = (unsigned*)(ws + OFF_HIST2);
  unsigned* cnt   = (unsigned*)(ws + OFF_CNT);
  unsigned* sel1  = (unsigned*)(ws + OFF_SEL1);
  unsigned* selK  = (unsigned*)(ws + OFF_SELK);
  unsigned long long* cand = (unsigned long long*)(ws + OFF_CAND);
  float* scoresTop = (float*)(ws + OFF_SCORES);
  float* boxesTop  = (float*)(ws + OFF_BOXES);
  float* vols      = (float*)(ws + OFF_VOLS);
  unsigned* maskp  = (unsigned*)(ws + OFF_MASK);

  const int ZN = (int)(OFF_ZEND / 4);
  k_zero<<<(ZN + 255) / 256, 256, 0, stream>>>((unsigned*)ws, ZN);

  const int HB = (TOTAL + 255) / 256;
  k_hist_hi<<<HB, 256, 0, stream>>>(scores, hist1);
  k_sel1<<<B_, 1024, 0, stream>>>(hist1, sel1);
  k_hist_lo<<<HB, 256, 0, stream>>>(scores, sel1, hist2);
  k_sel2<<<B_, 1024, 0, stream>>>(hist2, sel1, selK);
  k_gather<<<HB, 256, 0, stream>>>(scores, selK, cnt, cand);
  k_sortprep<<<B_, 1024, 0, stream>>>(cand, bboxes, scoresTop, boxesTop, vols);
  k_iou<<<dim3(64, 128, B_), 32, 0, stream>>>(boxesTop, vols, maskp);
  k_nmsout<<<B_, 64, 0, stream>>>(maskp, scoresTop, boxesTop, out);
}